// Group_2439541424181
// MI455X (gfx1250) — compile-verified
//
#include <hip/hip_runtime.h>
#include <hip/hip_bf16.h>

// ---------------------------------------------------------------------------
// Point-cloud grouping (FPS + 32-NN gather) for MI455X (gfx1250, wave32).
//   Kernel 1: farthest-point sampling, one workgroup per batch, points in
//             registers, single barrier per serial step (parity buffers).
//   Kernel 2: KNN via V_WMMA_F32_16X16X4_F32 score tiles (loads batched ahead
//             of the WMMAs) + wave-level bitonic top-32, then gather/subtract.
// ---------------------------------------------------------------------------

#define B_   32
#define N_   8192
#define G_   512
#define M_   32

typedef float v2f __attribute__((ext_vector_type(2)));
typedef float v8f __attribute__((ext_vector_type(8)));

// ---------------------------------------------------------------------------
// Wave32 bitonic helpers: (dist, idx) pairs, strict order (d, then idx).
// ---------------------------------------------------------------------------
__device__ __forceinline__ void ce_pair(float& d, int& i, int j, bool up, int lane) {
  float od = __shfl_xor(d, j, 32);
  int   oi = __shfl_xor(i, j, 32);
  bool lowLane   = ((lane & j) == 0);
  bool otherLess = (od < d) || (od == d && oi < i);
  bool takeOther = (lowLane == up) ? otherLess : !otherLess;
  d = takeOther ? od : d;
  i = takeOther ? oi : i;
}

// Full bitonic sort of 32 lanes, ascending by (d, idx).
__device__ __forceinline__ void sort32_asc(float& d, int& i, int lane) {
  { bool up = ((lane & 2) == 0);
    ce_pair(d, i, 1, up, lane); }
  { bool up = ((lane & 4) == 0);
    ce_pair(d, i, 2, up, lane); ce_pair(d, i, 1, up, lane); }
  { bool up = ((lane & 8) == 0);
    ce_pair(d, i, 4, up, lane); ce_pair(d, i, 2, up, lane); ce_pair(d, i, 1, up, lane); }
  { bool up = ((lane & 16) == 0);
    ce_pair(d, i, 8, up, lane); ce_pair(d, i, 4, up, lane);
    ce_pair(d, i, 2, up, lane); ce_pair(d, i, 1, up, lane); }
  { ce_pair(d, i, 16, true, lane); ce_pair(d, i, 8, true, lane);
    ce_pair(d, i, 4,  true, lane); ce_pair(d, i, 2, true, lane);
    ce_pair(d, i, 1,  true, lane); }
}

// Merge a bitonic 32-sequence into ascending order.
__device__ __forceinline__ void merge32_asc(float& d, int& i, int lane) {
  ce_pair(d, i, 16, true, lane); ce_pair(d, i, 8, true, lane);
  ce_pair(d, i, 4,  true, lane); ce_pair(d, i, 2, true, lane);
  ce_pair(d, i, 1,  true, lane);
}

// ---------------------------------------------------------------------------
// Kernel 1: FPS. One workgroup (1024 thr = 32 waves) per batch.
// Owned points + running min-dist in registers; LDS keeps an SoA copy only
// for the winner-coordinate broadcast. One barrier per serial step via
// parity-double-buffered wave-winner arrays (all waves reduce redundantly).
// ---------------------------------------------------------------------------
#define FPS_THREADS 1024
#define FPS_PER_THR (N_ / FPS_THREADS)   // 8

__global__ void fps_kernel(const float* __restrict__ xyz,
                           float* __restrict__ out_center) {
  extern __shared__ float sm[];
  float* sx   = sm;                       // [N]
  float* sy   = sm + N_;                  // [N]
  float* sz   = sm + 2 * N_;              // [N]
  float* rval = sm + 3 * N_;              // [2][32] parity buffers
  int*   ridx = (int*)(rval + 64);        // [2][32]

  const int b    = blockIdx.x;
  const int tid  = threadIdx.x;
  const int lane = tid & 31;
  const int wid  = tid >> 5;
  const float* p = xyz + (size_t)b * N_ * 3;

  // Load cloud: LDS copy (broadcast source) + private registers (hot data).
  float px[FPS_PER_THR], py[FPS_PER_THR], pz[FPS_PER_THR], md[FPS_PER_THR];
#pragma unroll
  for (int i = 0; i < FPS_PER_THR; ++i) {
    const int n = tid + i * FPS_THREADS;          // conflict-free stride
    const float x = p[n * 3 + 0];
    const float y = p[n * 3 + 1];
    const float z = p[n * 3 + 2];
    px[i] = x; py[i] = y; pz[i] = z; md[i] = 1e10f;
    sx[n] = x; sy[n] = y; sz[n] = z;
  }
  __syncthreads();

  int cur = 0;
  for (int j = 0; j < G_; ++j) {
    // record center j = pts[cur]  (scan emits idx before the update)
    if (tid == 0) {
      size_t o = ((size_t)b * G_ + j) * 3;
      out_center[o + 0] = sx[cur];
      out_center[o + 1] = sy[cur];
      out_center[o + 2] = sz[cur];
    }
    const float cx = sx[cur], cy = sy[cur], cz = sz[cur];  // LDS broadcast

    float bv = -1.0f; int bn = 0x7fffffff;
#pragma unroll
    for (int i = 0; i < FPS_PER_THR; ++i) {
      const float dx = px[i] - cx, dy = py[i] - cy, dz = pz[i] - cz;
      const float d  = dx * dx + dy * dy + dz * dz;
      const float m  = md[i] < d ? md[i] : d;
      md[i] = m;
      if (m > bv) { bv = m; bn = tid + i * FPS_THREADS; }  // strict > => first idx
    }
    // wave argmax, tie -> smaller index (matches jnp.argmax)
#pragma unroll
    for (int off = 16; off >= 1; off >>= 1) {
      float ov = __shfl_xor(bv, off, 32);
      int   on = __shfl_xor(bn, off, 32);
      if (ov > bv || (ov == bv && on < bn)) { bv = ov; bn = on; }
    }
    const int par = j & 1;
    if (lane == 0) { rval[par * 32 + wid] = bv; ridx[par * 32 + wid] = bn; }
    __syncthreads();
    // every wave reduces the 32 wave-winners redundantly (no 2nd barrier:
    // next iteration writes the other parity buffer)
    float v = rval[par * 32 + lane];
    int   n = ridx[par * 32 + lane];
#pragma unroll
    for (int off = 16; off >= 1; off >>= 1) {
      float ov = __shfl_xor(v, off, 32);
      int   on = __shfl_xor(n, off, 32);
      if (ov > v || (ov == v && on < n)) { v = ov; n = on; }
    }
    cur = n;                                       // identical in all lanes
  }
}

// ---------------------------------------------------------------------------
// Kernel 2: KNN + gather. One workgroup (512 thr = 16 waves) per
// (batch, 16-center tile). Scores via WMMA f32 16x16x4 with the
// [c,1] x [-2x, |x|^2] augmentation (per-row +|c|^2 dropped: rank-invariant).
// ---------------------------------------------------------------------------
#define KNN_WAVES 16
#define CHUNK     2048
#define ROWSTRIDE 2052                   // 2048 + 4: de-conflict 64 LDS banks
#define TILES_PER_WAVE (CHUNK / (16 * KNN_WAVES))   // 8

__global__ void knn_kernel(const float* __restrict__ xyz,
                           const float* __restrict__ centers,
                           float* __restrict__ out_nei) {
  extern __shared__ float sd2[];         // [16][ROWSTRIDE]

  const int b    = blockIdx.x;
  const int g0   = blockIdx.y * 16;
  const int tid  = threadIdx.x;
  const int lane = tid & 31;
  const int w    = tid >> 5;             // wave id == center row owned
  const float* pts = xyz + (size_t)b * N_ * 3;
  const float* cen = centers + ((size_t)b * G_ + g0) * 3;

  // A matrix (16 centers x K=4): lanes 0-15 hold {K0=cx, K1=cy},
  // lanes 16-31 hold {K2=cz, K3=1}.
  const int am = lane & 15;
  const float cAx = cen[am * 3 + 0];
  const float cAy = cen[am * 3 + 1];
  const float cAz = cen[am * 3 + 2];
  v2f A;
  A.x = (lane < 16) ? cAx : cAz;
  A.y = (lane < 16) ? cAy : 1.0f;

  // running top-32: lane l holds l-th smallest (score, idx)
  float bd  = 3.402823466e38f;
  int   bi  = 0;
  float thr = 3.402823466e38f;

  for (int cb = 0; cb < N_; cb += CHUNK) {
    // ---- load pass: batch all 8 point tiles (8 back-to-back b96 loads) --
    float qx[TILES_PER_WAVE], qy[TILES_PER_WAVE], qz[TILES_PER_WAVE];
#pragma unroll
    for (int t = 0; t < TILES_PER_WAVE; ++t) {
      const int n = cb + (w * TILES_PER_WAVE + t) * 16 + (lane & 15);
      qx[t] = pts[n * 3 + 0];
      qy[t] = pts[n * 3 + 1];
      qz[t] = pts[n * 3 + 2];
    }
    // ---- WMMA pass: 8 16x16 score tiles per wave ------------------------
#pragma unroll
    for (int t = 0; t < TILES_PER_WAVE; ++t) {
      const float x2 = qx[t] * qx[t] + qy[t] * qy[t] + qz[t] * qz[t];
      // B matrix (K=4 x 16 points): lanes 0-15 {K0=-2x, K1=-2y},
      // lanes 16-31 {K2=-2z, K3=|x|^2}
      v2f Bv;
      Bv.x = (lane < 16) ? (-2.0f * qx[t]) : (-2.0f * qz[t]);
      Bv.y = (lane < 16) ? (-2.0f * qy[t]) : x2;

      v8f acc = {};
      acc = __builtin_amdgcn_wmma_f32_16x16x4_f32(
          /*neg_a=*/false, A, /*neg_b=*/false, Bv,
          /*c_mod=*/(short)0, acc, /*reuse_a=*/false, /*reuse_b=*/false);

      const int col   = (w * TILES_PER_WAVE + t) * 16 + (lane & 15);
      const int rbase = (lane < 16) ? 0 : 8;          // C/D layout: M = r + 8*half
#pragma unroll
      for (int r = 0; r < 8; ++r)
        sd2[(rbase + r) * ROWSTRIDE + col] = acc[r];
    }
    __syncthreads();

    // ---- selection phase: wave w scans its row, 32 candidates/step ------
    for (int c = 0; c < CHUNK / 32; ++c) {
      float cd = sd2[w * ROWSTRIDE + c * 32 + lane];
      int   ci = cb + c * 32 + lane;
      if (__ballot(cd < thr) == 0ull) continue;       // threshold prune

      sort32_asc(cd, ci, lane);                       // candidates ascending
      float rd = __shfl_xor(cd, 31, 32);              // reversed (descending)
      int   ri = __shfl_xor(ci, 31, 32);
      bool takeOther = (rd < bd) || (rd == bd && ri < bi);
      bd = takeOther ? rd : bd;                       // elementwise min -> bitonic
      bi = takeOther ? ri : bi;
      merge32_asc(bd, bi, lane);                      // re-sort ascending
      thr = __shfl(bd, 31, 32);
    }
    __syncthreads();
  }

  // ---- gather neighborhood, subtract center ----------------------------
  const float cwx = cen[w * 3 + 0];
  const float cwy = cen[w * 3 + 1];
  const float cwz = cen[w * 3 + 2];
  const int   gi  = bi;
  const float px = pts[gi * 3 + 0];
  const float py = pts[gi * 3 + 1];
  const float pz = pts[gi * 3 + 2];
  const size_t o = (((size_t)b * G_ + g0 + w) * M_ + lane) * 3;
  out_nei[o + 0] = px - cwx;
  out_nei[o + 1] = py - cwy;
  out_nei[o + 2] = pz - cwz;
}

// ---------------------------------------------------------------------------
extern "C" void kernel_launch(void* const* d_in, const int* in_sizes, int n_in,
                              void* d_out, int out_size, void* d_ws, size_t ws_size,
                              hipStream_t stream) {
  (void)in_sizes; (void)n_in; (void)out_size; (void)d_ws; (void)ws_size;
  const float* xyz = (const float*)d_in[0];
  float* out     = (float*)d_out;
  float* out_nei = out;                                 // [B,G,M,3]
  float* out_cen = out + (size_t)B_ * G_ * M_ * 3;      // [B,G,3]

  const size_t fps_lds = (size_t)(3 * N_ + 64 + 64) * sizeof(float);
  fps_kernel<<<B_, FPS_THREADS, fps_lds, stream>>>(xyz, out_cen);

  dim3 grid2(B_, G_ / 16);
  const size_t knn_lds = (size_t)16 * ROWSTRIDE * sizeof(float);
  knn_kernel<<<grid2, 32 * KNN_WAVES, knn_lds, stream>>>(xyz, out_cen, out_nei);
}